// MY_CTC_LOSS_19722489823856
// MI455X (gfx1250) — compile-verified
//
#include <hip/hip_runtime.h>
#include <cstdint>
#include <math.h>

// ---------------------------------------------------------------------------
// CTC loss forward (reduction='none') for T=1024, B=32, V=1024, L=128.
//
// MI455X reasoning: total traffic = T*B*V*4 = 128 MB (~5.5us @ 23.3 TB/s);
// compute is negligible. The kernel is a sequential scan over T, so the
// implementation streams each log_probs[t,b,:] row (4 KB) into LDS with the
// CDNA5 async-to-LDS path (ASYNCcnt), 7 rows deep, and runs the 258-wide
// alpha recurrence out of LDS with one workgroup barrier per frame.
// ---------------------------------------------------------------------------

#define NEGF (-3.402823466e+38f)   // torch.finfo(float32).min, matches reference

constexpr int CT_V = 1024;          // vocab
constexpr int CT_L = 128;           // max target length
constexpr int CT_S = 2 * CT_L + 2;  // extended lattice size = 258
constexpr int NB   = 8;             // LDS row ring depth (prefetch depth NB-1)
constexpr int NTHREADS = 288;       // >= max(CT_S, 256); 9 waves

__global__ __launch_bounds__(NTHREADS, 1)
void ctc_loss_fwd_kernel(const float* __restrict__ log_probs,   // [T,B,V]
                         const int*   __restrict__ targets,     // [B,L]
                         const int*   __restrict__ input_len,   // [B]
                         const int*   __restrict__ target_len,  // [B]
                         float* __restrict__ out,               // [B]
                         int T, int B)
{
    __shared__ float rows[NB][CT_V];       // 32 KB ring of log-prob rows
    __shared__ float alphaBuf[2][CT_S];    // double-buffered alpha
    __shared__ int   labels[CT_S];         // extended (blank-interleaved) labels
    __shared__ float fin[2];               // terminal alphas

    const int b    = blockIdx.x;
    const int tid  = threadIdx.x;
    const int ilen = input_len[b];
    const int tlen = target_len[b];

    // ---- extended label sequence: [blank, t0, blank, t1, ..., blank, t_L] --
    // t_ext = concat(targets, targets[:, :1])  (reference appends first label)
    if (tid < CT_S) {
        int lab = 0;
        if (tid & 1) {
            int k = tid >> 1;                       // 0..L
            lab = targets[b * CT_L + (k < CT_L ? k : 0)];
        }
        labels[tid] = lab;
    }
    __syncthreads();

    int  myLab  = 0;
    bool myDiff = false;
    if (tid < CT_S) {
        myLab  = labels[tid];
        myDiff = (tid >= 2) && (labels[tid] != labels[tid - 2]);
    }

    const size_t rowStride = (size_t)B * CT_V;

    // ---- prologue: issue async prefetch for rows 0..NB-2 ------------------
    for (int r = 0; r < NB - 1; ++r) {
        if (r < ilen && tid < 256) {
            const float* gp = log_probs + (size_t)r * rowStride
                                        + (size_t)b * CT_V + (size_t)tid * 4;
            uint32_t laddr = (uint32_t)(uintptr_t)(&rows[r][tid * 4]);
            asm volatile("global_load_async_to_lds_b128 %0, %1, off"
                         :: "v"(laddr),
                            "v"((unsigned long long)(uintptr_t)gp)
                         : "memory");
        }
    }

    const int s1 = 2 * tlen - 1;
    const int s2 = 2 * tlen;

    // ---- alpha recursion over frames --------------------------------------
    for (int t = 0; t < ilen; ++t) {
        // own wave's slice of row t has landed (loads complete in order)
        asm volatile("s_wait_asynccnt 6" ::: "memory");   // <= NB-2 outstanding
        // all waves' slices of row t landed; slot (t-1)%NB now reusable
        __syncthreads();

        // prefetch row t+NB-1 into slot (t-1)%NB (consumed at iter t-1,
        // protected by the barrier above)
        {
            const int r = t + NB - 1;
            if (r < ilen && tid < 256) {
                const float* gp = log_probs + (size_t)r * rowStride
                                            + (size_t)b * CT_V + (size_t)tid * 4;
                uint32_t laddr = (uint32_t)(uintptr_t)(&rows[r % NB][tid * 4]);
                asm volatile("global_load_async_to_lds_b128 %0, %1, off"
                             :: "v"(laddr),
                                "v"((unsigned long long)(uintptr_t)gp)
                             : "memory");
            }
        }

        if (tid < CT_S) {
            const float* row = rows[t % NB];
            const float  lp  = row[myLab];            // LDS gather
            float a;
            if (t == 0) {
                // alpha0: s=0 -> lp(blank), s=1 -> lp(label1), else NEG
                a = (tid < 2) ? lp : NEGF;
            } else {
                const float* ap = alphaBuf[(t - 1) & 1];
                const float a1 = ap[tid];
                const float a2 = (tid >= 1) ? ap[tid - 1] : NEGF;
                const float a3 = myDiff     ? ap[tid - 2] : NEGF;
                const float m  = fmaxf(a1, fmaxf(a2, a3));
                // exp(NEG - m) == 0 for normal m; all-NEG case rounds back to NEG
                const float ssum = expf(a1 - m) + expf(a2 - m) + expf(a3 - m);
                a = lp + m + logf(ssum);
            }
            alphaBuf[t & 1][tid] = a;

            if (t == ilen - 1) {                      // capture terminal states
                if (tid == s1) fin[0] = a;
                if (tid == s2) fin[1] = a;
            }
        }
    }

    __syncthreads();

    if (tid == 0) {
        const float f0 = fin[0];
        const float f1 = fin[1];
        const float m  = fmaxf(f0, f1);
        out[b] = -(m + logf(expf(f0 - m) + expf(f1 - m)));
    }
}

// ---------------------------------------------------------------------------
extern "C" void kernel_launch(void* const* d_in, const int* in_sizes, int n_in,
                              void* d_out, int out_size, void* d_ws, size_t ws_size,
                              hipStream_t stream)
{
    const float* log_probs  = (const float*)d_in[0];   // [T,B,V] f32
    const int*   targets    = (const int*)  d_in[1];   // [B,L]
    const int*   input_len  = (const int*)  d_in[2];   // [B]
    const int*   target_len = (const int*)  d_in[3];   // [B]
    float*       out        = (float*)d_out;           // [B]

    const int B = in_sizes[2];
    const int T = (int)((long long)in_sizes[0] / ((long long)B * CT_V));

    ctc_loss_fwd_kernel<<<B, NTHREADS, 0, stream>>>(
        log_probs, targets, input_len, target_len, out, T, B);
}